// KnowledgeBaseLookup_47330539602109
// MI455X (gfx1250) — compile-verified
//
#include <hip/hip_runtime.h>

// ---------------------------------------------------------------------------
// Fused: y[b,t,:] = sum_k weights[b,t,k] * KB[idx[b,t,k],:]   (gather, f32)
//        out      = y @ w_out^T + b_out                        (WMMA f32)
// B=4 T=2048 K=16 C=262144 D=256 E=512  -> BT=8192 tokens
// ---------------------------------------------------------------------------

typedef __attribute__((ext_vector_type(2))) float v2f;
typedef __attribute__((ext_vector_type(8))) float v8f;

#define KSAMP 16
#define D_DIM 256
#define E_DIM 512
#define M_TILE 16          // tokens per workgroup
#define WAVES 8            // 256 threads, wave32
#define N_PER_WAVE 64      // 512 cols / 8 waves
#define NTILES 4           // 64 / 16

__global__ __launch_bounds__(256) void kb_lookup_proj_kernel(
    const float* __restrict__ weights,   // (BT, 16)
    const int*   __restrict__ indexes,   // (BT, 16)
    const float* __restrict__ kb,        // (C, 256)
    const float* __restrict__ w_out,     // (512, 256) row-major (E, D)
    const float* __restrict__ b_out,     // (512)
    float*       __restrict__ out)       // (BT, 512)
{
    __shared__ float y[M_TILE][D_DIM];   // 16 KB

    const int tid   = threadIdx.x;       // 0..255
    const int mbase = blockIdx.x * M_TILE;

    // ---------------- Phase 1: gather + weighted reduction into LDS --------
    // For each of the 16 tokens, the 16 (index, weight) pairs are uniform
    // across the workgroup -> scalar loads; 256 lanes each fetch one f32 of
    // the 1KB knowledge-base row (coalesced global_load_b32).
    for (int m = 0; m < M_TILE; ++m) {
        const int tok = mbase + m;
        const float* wrow = weights + (size_t)tok * KSAMP;
        const int*   irow = indexes + (size_t)tok * KSAMP;
        float acc = 0.0f;
        #pragma unroll
        for (int k = 0; k < KSAMP; ++k) {
            const long  idx = (long)irow[k];          // uniform -> SGPR
            const float w   = wrow[k];                // uniform -> SGPR
            acc = fmaf(w, kb[idx * (long)D_DIM + tid], acc);
        }
        y[m][tid] = acc;
    }
    __syncthreads();

    // ---------------- Phase 2: out[16 x 512] tile via V_WMMA_F32_16X16X4_F32
    const int wave  = tid >> 5;          // 0..7 -> column slab
    const int lane  = tid & 31;
    const int half  = lane >> 4;         // K-half selector (ISA 16x4 layout)
    const int lrow  = lane & 15;         // M (for A) / N (for B,C,D)
    const int nbase = wave * N_PER_WAVE;

    // Accumulators initialised with the bias (same bias for every row of a
    // column -> broadcast across the 8 C VGPRs).
    v8f acc[NTILES];
    const float* brow[NTILES];
    #pragma unroll
    for (int t = 0; t < NTILES; ++t) {
        const int col = nbase + t * 16 + lrow;
        const float bias = b_out[col];
        acc[t] = (v8f){bias, bias, bias, bias, bias, bias, bias, bias};
        brow[t] = w_out + (size_t)col * D_DIM;   // B(k,n) = w_out[n][k]
    }

    // K loop: 256 / 4 = 64 steps, 4 wmma each (one per 16-col tile).
    for (int k0 = 0; k0 < D_DIM; k0 += 4) {
        const int ko = k0 + 2 * half;            // this lane's K pair
        // A fragment: row = lrow (token), k = ko, ko+1  (8B-aligned LDS read)
        const v2f a = *(const v2f*)&y[lrow][ko];
        #pragma unroll
        for (int t = 0; t < NTILES; ++t) {
            // B fragment: column = tile col, k = ko, ko+1 (8B global read, L2 hot)
            const v2f b = *(const v2f*)(brow[t] + ko);
            acc[t] = __builtin_amdgcn_wmma_f32_16x16x4_f32(
                /*neg_a=*/false, a, /*neg_b=*/false, b,
                /*c_mod=*/(short)0, acc[t],
                /*reuse_a=*/false, /*reuse_b=*/false);
        }
    }

    // ---------------- Store: C/D layout — VGPR r holds row (r + 8*half) ----
    #pragma unroll
    for (int t = 0; t < NTILES; ++t) {
        const int col = nbase + t * 16 + lrow;
        #pragma unroll
        for (int r = 0; r < 8; ++r) {
            const int row = mbase + r + 8 * half;
            out[(size_t)row * E_DIM + col] = acc[t][r];
        }
    }
}

extern "C" void kernel_launch(void* const* d_in, const int* in_sizes, int n_in,
                              void* d_out, int out_size, void* d_ws, size_t ws_size,
                              hipStream_t stream) {
    const float* weights = (const float*)d_in[0];   // (4,2048,16)
    const int*   indexes = (const int*)  d_in[1];   // (4,2048,16)
    const float* kb      = (const float*)d_in[2];   // (262144,256)
    const float* w_out   = (const float*)d_in[3];   // (512,256)
    const float* b_out   = (const float*)d_in[4];   // (512,)
    float*       out     = (float*)d_out;           // (4,2048,512)

    const int BT = 4 * 2048;                        // 8192 tokens
    dim3 grid(BT / M_TILE);                         // 512 workgroups
    dim3 block(256);                                // 8 wave32s
    kb_lookup_proj_kernel<<<grid, block, 0, stream>>>(
        weights, indexes, kb, w_out, b_out, out);
}